// Actor_CPG_48146583388367
// MI455X (gfx1250) — compile-verified
//
#include <hip/hip_runtime.h>

typedef __attribute__((ext_vector_type(2))) float v2f;
typedef __attribute__((ext_vector_type(8))) float v8f;

#define NN 12
#define NP 16              // padded feature dim (zero rows 12..15)
#define PP 24
#define OBSD 60
#define ROWS_PER_BLOCK 128
#define ROWS_PER_WAVE 16
#define DT_STEP 0.002f
#define TWO_PI 6.28318530717958647692f

static __device__ __forceinline__ v8f wmma4(v2f a, v2f b, v8f c) {
    return __builtin_amdgcn_wmma_f32_16x16x4_f32(false, a, false, b, (short)0, c,
                                                 false, false);
}

__launch_bounds__(256)
__global__ void cpg_fused_kernel(
    const float* __restrict__ obs,
    const float* __restrict__ theta_old,
    const float* __restrict__ theta_dot_old,
    const float* __restrict__ theta_dot_dot_old,
    const float* __restrict__ r_old,
    const float* __restrict__ r_dot_old,
    const float* __restrict__ r_dot_dot_old,
    const float* __restrict__ v_short,
    const float* __restrict__ sym,
    const float* __restrict__ fixedv,
    const float* __restrict__ Wd,
    const float* __restrict__ Ws,
    const float* __restrict__ Cd,
    const float* __restrict__ Od,
    const float* __restrict__ Wm,
    const float* __restrict__ Fi,
    const float* __restrict__ Am,
    const float* __restrict__ Cr,
    const float* __restrict__ Orr,
    const float* __restrict__ Lambda,
    const float* __restrict__ Lambda_T,
    const float* __restrict__ SIGMA,
    const float* __restrict__ Dm,
    float* __restrict__ out,
    int Btot)
{
    __shared__ float s_obs[ROWS_PER_BLOCK * OBSD];   // 30720 B
    __shared__ float s_Wd[NP * OBSD];                // padded: rows 12..15 zero
    __shared__ float s_Ws[NP * OBSD];
    __shared__ float s_D[NP * NN];
    __shared__ float s_SIG[NP * NN];
    __shared__ float s_Lam[NP * NN];
    __shared__ float s_LC[NP * NN];                  // Lambda - Lambda_T
    __shared__ float s_v[PP];
    __shared__ float s_vec[7 * NN];                  // Cdv,Odv,Wv,Fiv,Av,Crv,Orv
    __shared__ float s_scr[8 * 256];                 // per-wave bounce tile

    const int tid = threadIdx.x;
    const int blockRow = blockIdx.x * ROWS_PER_BLOCK;

    // ---- Phase 1: cooperative staging (coalesced, vectorized) ----
    {
        const float4* src = (const float4*)(obs + (size_t)blockRow * OBSD);
        float4* dst = (float4*)s_obs;
        const int nvec = ROWS_PER_BLOCK * OBSD / 4;            // 1920
        const int limit = (Btot * OBSD - blockRow * OBSD) / 4; // full for exact grid
        for (int i = tid; i < nvec; i += 256)
            dst[i] = (i < limit) ? src[i] : float4{0.f, 0.f, 0.f, 0.f};
    }
    for (int i = tid; i < NP * OBSD; i += 256) {
        float wd = 0.f, ws = 0.f;
        if (i < NN * OBSD) { wd = Wd[i]; ws = Ws[i]; }
        s_Wd[i] = wd; s_Ws[i] = ws;
    }
    for (int i = tid; i < NP * NN; i += 256) {
        float d = 0.f, sg = 0.f, lm = 0.f, lc = 0.f;
        if (i < NN * NN) {
            d = Dm[i]; sg = SIGMA[i];
            lm = Lambda[i]; lc = Lambda[i] - Lambda_T[i];
        }
        s_D[i] = d; s_SIG[i] = sg; s_Lam[i] = lm; s_LC[i] = lc;
    }
    if (tid < PP) {
        float acc = fixedv[tid];
        for (int i = 0; i < 12; ++i) acc += sym[tid * 12 + i] * v_short[i];
        s_v[tid] = acc;
    }
    __syncthreads();
    if (tid < 7 * NN) {
        const int vi = tid / NN, n = tid % NN;
        const float* Mp = (vi == 0) ? Cd : (vi == 1) ? Od : (vi == 2) ? Wm
                        : (vi == 3) ? Fi : (vi == 4) ? Am : (vi == 5) ? Cr : Orr;
        float acc = 0.f;
        for (int p = 0; p < PP; ++p) acc += Mp[n * PP + p] * s_v[p];
        s_vec[tid] = acc;
    }
    __syncthreads();

    // ---- Phase 2: per-wave WMMA pipeline ----
    const int wave = tid >> 5;
    const int lane = tid & 31;
    const int L = lane & 15;          // feature / M index within half-wave
    const int h = lane >> 4;          // which K/M half
    const int wrow = wave * ROWS_PER_WAVE;
    const int gRow = blockRow + wrow;
    float* scr = &s_scr[wave * 256];

    // d = obs @ Wd.T ; sigma_N = obs @ Ws.T   (K = 60, 15 chunks of 4)
    v8f acc_d = {};
    v8f acc_s = {};
#pragma unroll
    for (int c = 0; c < 15; ++c) {
        const int k = 4 * c + 2 * h;
        v2f a;  a.x  = s_obs[(wrow + L) * OBSD + k];
                a.y  = s_obs[(wrow + L) * OBSD + k + 1];
        v2f bd; bd.x = s_Wd[L * OBSD + k];
                bd.y = s_Wd[L * OBSD + k + 1];
        v2f bs; bs.x = s_Ws[L * OBSD + k];
                bs.y = s_Ws[L * OBSD + k + 1];
        acc_d = wmma4(a, bd, acc_d);
        acc_s = wmma4(a, bs, acc_s);
    }

    // Dd = d @ D.T : bounce d (C-layout) -> scratch -> A-layout, K=12 (3 chunks)
#pragma unroll
    for (int j = 0; j < 8; ++j) scr[(j + 8 * h) * 16 + L] = acc_d[j];
    v8f acc_Dd = {};
#pragma unroll
    for (int c = 0; c < 3; ++c) {
        const int k = 4 * c + 2 * h;
        v2f a; a.x = scr[L * 16 + k];    a.y = scr[L * 16 + k + 1];
        v2f b; b.x = s_D[L * NN + k];    b.y = s_D[L * NN + k + 1];
        acc_Dd = wmma4(a, b, acc_Dd);
    }

    // sigN = sigma_N @ SIGMA.T
#pragma unroll
    for (int j = 0; j < 8; ++j) scr[(j + 8 * h) * 16 + L] = acc_s[j];
    v8f acc_sg = {};
#pragma unroll
    for (int c = 0; c < 3; ++c) {
        const int k = 4 * c + 2 * h;
        v2f a; a.x = scr[L * 16 + k];    a.y = scr[L * 16 + k + 1];
        v2f b; b.x = s_SIG[L * NN + k];  b.y = s_SIG[L * NN + k + 1];
        acc_sg = wmma4(a, b, acc_sg);
    }

    // lam_r = r_old @ Lambda.T  (stage 16x12 tile = 192 contiguous floats)
    for (int t = lane; t < ROWS_PER_WAVE * NN; t += 32)
        scr[t] = r_old[gRow * NN + t];
    v8f acc_lr = {};
#pragma unroll
    for (int c = 0; c < 3; ++c) {
        const int k = 4 * c + 2 * h;
        v2f a; a.x = scr[L * NN + k];    a.y = scr[L * NN + k + 1];
        v2f b; b.x = s_Lam[L * NN + k];  b.y = s_Lam[L * NN + k + 1];
        acc_lr = wmma4(a, b, acc_lr);
    }

    // lam_th = theta_old @ (Lambda - Lambda_T).T
    for (int t = lane; t < ROWS_PER_WAVE * NN; t += 32)
        scr[t] = theta_old[gRow * NN + t];
    v8f acc_lt = {};
#pragma unroll
    for (int c = 0; c < 3; ++c) {
        const int k = 4 * c + 2 * h;
        v2f a; a.x = scr[L * NN + k];    a.y = scr[L * NN + k + 1];
        v2f b; b.x = s_LC[L * NN + k];   b.y = s_LC[L * NN + k + 1];
        acc_lt = wmma4(a, b, acc_lt);
    }

    // ---- Phase 3: elementwise Hopf dynamics + trapezoidal integration ----
    const int strideO = Btot * NN;
#pragma unroll
    for (int j = 0; j < 8; ++j) {
        const int row = gRow + j + 8 * h;
        if (L < NN && row < Btot) {
            const int base = row * NN + L;
            const float th   = theta_old[base];
            const float thd  = theta_dot_old[base];
            const float thdd = theta_dot_dot_old[base];
            const float r    = r_old[base];
            const float rd   = r_dot_old[base];
            const float rdd  = r_dot_dot_old[base];
            const float Ddv  = acc_Dd[j];
            const float lamr = acc_lr[j];
            const float lamt = acc_lt[j];
            const float sg   = acc_sg[j];
            const float cdv = s_vec[0 * NN + L], odv = s_vec[1 * NN + L];
            const float wv  = s_vec[2 * NN + L], fiv = s_vec[3 * NN + L];
            const float av  = s_vec[4 * NN + L], crv = s_vec[5 * NN + L];
            const float orv = s_vec[6 * NN + L];

            const float ct = __cosf(th), st = __sinf(th);
            const float theta_dot = TWO_PI * (cdv * Ddv + odv)
                                  + wv * lamr * __sinf(lamt - fiv)
                                  - sg * ct;
            const float r_dd = av * (av * 0.25f * (crv * Ddv + orv - r) - rd);

            const float x     = r * ct;
            const float x_dot = rd * ct - r * st * thd;
            const float x_dd  = rdd * ct - 2.f * rd * st * thd
                              - r * (ct * thd * thd + st * thdd);
            const float theta_new = th + (theta_dot + thd) * (DT_STEP * 0.5f);
            const float r_dot_new = rd + (rdd + r_dd) * (DT_STEP * 0.5f);
            const float r_new     = r + (rd + r_dot_new) * (DT_STEP * 0.5f);
            const float theta_dd  = (theta_dot - thd) * (1.0f / DT_STEP);

            out[0 * strideO + base] = x;
            out[1 * strideO + base] = x_dot;
            out[2 * strideO + base] = x_dd;
            out[3 * strideO + base] = theta_new;
            out[4 * strideO + base] = theta_dot;
            out[5 * strideO + base] = theta_dd;
            out[6 * strideO + base] = r_new;
            out[7 * strideO + base] = r_dot_new;
            out[8 * strideO + base] = r_dd;
        }
    }
}

extern "C" void kernel_launch(void* const* d_in, const int* in_sizes, int n_in,
                              void* d_out, int out_size, void* d_ws, size_t ws_size,
                              hipStream_t stream) {
    const float* obs    = (const float*)d_in[0];
    const float* th     = (const float*)d_in[1];
    const float* thd    = (const float*)d_in[2];
    const float* thdd   = (const float*)d_in[3];
    const float* r      = (const float*)d_in[4];
    const float* rd     = (const float*)d_in[5];
    const float* rdd    = (const float*)d_in[6];
    const float* vshort = (const float*)d_in[7];
    const float* sym    = (const float*)d_in[8];
    const float* fixedv = (const float*)d_in[9];
    const float* Wd     = (const float*)d_in[10];
    const float* Ws     = (const float*)d_in[11];
    const float* Cd     = (const float*)d_in[12];
    const float* Od     = (const float*)d_in[13];
    const float* Wm     = (const float*)d_in[14];
    const float* Fi     = (const float*)d_in[15];
    const float* Am     = (const float*)d_in[16];
    const float* Cr     = (const float*)d_in[17];
    const float* Orr    = (const float*)d_in[18];
    const float* Lam    = (const float*)d_in[19];
    const float* LamT   = (const float*)d_in[20];
    const float* SIG    = (const float*)d_in[21];
    const float* Dm     = (const float*)d_in[22];
    float* out = (float*)d_out;

    const int Btot = in_sizes[0] / OBSD;               // 524288
    const int grid = (Btot + ROWS_PER_BLOCK - 1) / ROWS_PER_BLOCK;  // 4096

    cpg_fused_kernel<<<grid, 256, 0, stream>>>(
        obs, th, thd, thdd, r, rd, rdd, vshort, sym, fixedv,
        Wd, Ws, Cd, Od, Wm, Fi, Am, Cr, Orr, Lam, LamT, SIG, Dm,
        out, Btot);
}